// computeMaskedOutput_47382079209762
// MI455X (gfx1250) — compile-verified
//
#include <hip/hip_runtime.h>
#include <hip/hip_bf16.h>
#include <stdint.h>

// Problem constants (from reference): B=128, H=W=28, C=128
#define BATCH    128
#define HW       784                    // 28*28
#define CHANNELS 128
#define NELEM    (128L * 784L * 128L)   // 12,845,056 per output tensor

#define TC      16    // channels per tile in kernel 2
#define TSTRIDE 788   // 784 padded +4 floats: bank-conflict-free, keeps 16B alignment

#define NCHUNK   16   // hw chunks in kernel 1
#define CHUNK_HW 49   // 784 / 16

// ---------------------------------------------------------------------------
// Kernel 1: per-(b,c) spatial argmax over 784 positions.
// 512 threads (16 wave32) per block, one block per batch.
//   lane group: (t & 31) -> 4 consecutive channels via one float4 load
//   chunk:      (t >> 5) -> one of 16 contiguous hw ranges of 49
// Each wave load instruction moves 512 contiguous bytes; 4 independent
// compare chains per thread let loads run far ahead of the dependent chain.
// Chunks are merged in ascending hw order with strict '>' so first-occurrence
// semantics match jnp.argmax exactly.
// ---------------------------------------------------------------------------
__global__ void __launch_bounds__(512)
spatial_argmax_kernel(const float* __restrict__ x, int* __restrict__ star) {
    __shared__ __align__(16) float pmax[NCHUNK * CHANNELS];
    __shared__ __align__(16) int   pidx[NCHUNK * CHANNELS];

    const int b     = blockIdx.x;
    const int t     = threadIdx.x;
    const int cg    = (t & 31) * 4;   // channel base for this thread
    const int chunk = t >> 5;         // [0,16)
    const int hw0   = chunk * CHUNK_HW;

    const float4* p = (const float4*)(x + (long)b * HW * CHANNELS
                                        + (long)hw0 * CHANNELS + cg);
    // hw step in float4 units: CHANNELS/4 = 32
    float4 v = p[0];
    float b0 = v.x, b1 = v.y, b2 = v.z, b3 = v.w;
    int   i0 = hw0, i1 = hw0, i2 = hw0, i3 = hw0;

    #pragma unroll 8
    for (int k = 1; k < CHUNK_HW; ++k) {
        float4 u  = p[(long)k * (CHANNELS / 4)];
        int    hw = hw0 + k;
        if (u.x > b0) { b0 = u.x; i0 = hw; }
        if (u.y > b1) { b1 = u.y; i1 = hw; }
        if (u.z > b2) { b2 = u.z; i2 = hw; }
        if (u.w > b3) { b3 = u.w; i3 = hw; }
    }

    const int base = chunk * CHANNELS + cg;
    pmax[base + 0] = b0;  pmax[base + 1] = b1;
    pmax[base + 2] = b2;  pmax[base + 3] = b3;
    pidx[base + 0] = i0;  pidx[base + 1] = i1;
    pidx[base + 2] = i2;  pidx[base + 3] = i3;
    __syncthreads();

    if (t < CHANNELS) {
        float best = pmax[t];
        int   bi   = pidx[t];
        #pragma unroll
        for (int k = 1; k < NCHUNK; ++k) {          // ascending hw order
            float vv = pmax[k * CHANNELS + t];
            int   ii = pidx[k * CHANNELS + t];
            if (vv > best) { best = vv; bi = ii; }  // strict > keeps first max
        }
        star[b * CHANNELS + t] = bi;
    }
}

// ---------------------------------------------------------------------------
// Kernel 2: gather template rows via async global->LDS DMA, transpose through
// LDS, fuse relu(x*t) and emit all three outputs in NHWC.
// grid = (8 channel tiles, 128 batches), block = 256 threads (8 wave32).
// ---------------------------------------------------------------------------
__global__ void __launch_bounds__(256)
template_gather_kernel(const float* __restrict__ x,
                       const float* __restrict__ t_p,
                       const int*   __restrict__ star,
                       float*       __restrict__ out) {
    __shared__ __align__(16) float s_tile[TC * TSTRIDE];  // ~49.3 KB
    __shared__ int s_star[TC];

    const int t  = threadIdx.x;
    const int c0 = blockIdx.x * TC;
    const int b  = blockIdx.y;

    if (t < TC) {
        s_star[t] = star[b * CHANNELS + c0 + t];
    }
    __syncthreads();

    // ---- Phase 1: async DMA of TC template rows (each 784 contiguous floats)
    // into LDS. 3136 x 16-byte transfers per block. All addresses 16B aligned:
    //   global: ((c*784 + s)*784 + 4*vec)*4  -> multiple of 16
    //   lds   : c_local*788*4 + vec*16       -> multiple of 16
    const uint32_t lds_base = (uint32_t)(uintptr_t)s_tile;
    const uint64_t tp_base  = (uint64_t)(uintptr_t)t_p;

    #pragma unroll
    for (int i = 0; i < 13; ++i) {
        int idx = t + i * 256;              // vec4 slot in [0, 3136)
        if (idx < TC * (HW / 4)) {
            int c_local = idx / (HW / 4);   // / 196
            int vec     = idx - c_local * (HW / 4);
            int s       = s_star[c_local];  // flat argmax h*W + w
            // t_p[c, h*, w*, :, :] starts at ((c*784) + s) * 784 floats
            long foff = ((long)(c0 + c_local) * HW + s) * HW + (long)vec * 4;
            uint64_t gaddr = tp_base + (uint64_t)foff * 4u;
            uint32_t laddr = lds_base + (uint32_t)(c_local * TSTRIDE + vec * 4) * 4u;
            asm volatile("global_load_async_to_lds_b128 %0, %1, off"
                         :: "v"(laddr), "v"(gaddr) : "memory");
        }
    }
    // Each wave drains its own ASYNCcnt, then the block barrier makes all
    // async LDS writes visible to every wave.
    asm volatile("s_wait_asynccnt 0x0" ::: "memory");
    __syncthreads();

    // ---- Phase 2: transposed, fused output. Lane mapping puts 16 consecutive
    // channels in consecutive lanes -> 64B contiguous segments per half-wave
    // for all global reads/writes. LDS reads are bank-conflict-free (stride
    // 788 floats => 20*c mod 64 distinct over the 16-channel tile).
    #pragma unroll 2
    for (int i = 0; i < (TC * HW) / 256; ++i) {   // 49 iterations
        int idx     = t + i * 256;                // [0, 12544)
        int c_local = idx & (TC - 1);
        int hw      = idx >> 4;                   // [0, 784)
        long oidx   = ((long)b * HW + hw) * CHANNELS + c0 + c_local;

        float xv = x[oidx];
        float tv = s_tile[c_local * TSTRIDE + hw];

        out[oidx]             = fmaxf(xv * tv, 0.0f);  // masked = relu(x * t)
        out[NELEM + oidx]     = xv;                    // passthrough x
        out[2 * NELEM + oidx] = tv;                    // templates (NHWC)
    }
}

// ---------------------------------------------------------------------------
extern "C" void kernel_launch(void* const* d_in, const int* in_sizes, int n_in,
                              void* d_out, int out_size, void* d_ws, size_t ws_size,
                              hipStream_t stream) {
    const float* x   = (const float*)d_in[0];   // [128,28,28,128] f32
    const float* t_p = (const float*)d_in[1];   // [128,28,28,28,28] f32
    float* out = (float*)d_out;                 // masked | x | templates, flat
    int* star = (int*)d_ws;                     // 128*128 ints = 64 KB scratch

    spatial_argmax_kernel<<<dim3(BATCH), dim3(512), 0, stream>>>(x, star);

    template_gather_kernel<<<dim3(CHANNELS / TC, BATCH), dim3(256), 0, stream>>>(
        x, t_p, star, out);
}